// VoxelGenerate_47863115546693
// MI455X (gfx1250) — compile-verified
//
#include <hip/hip_runtime.h>
#include <hip/hip_bf16.h>

// ---------------- problem constants (match reference) ----------------
#define GXC 512
#define GYC 512
#define GZC 40
#define GC  (GXC * GYC * GZC)          // 10,485,760 cells
#define VC  60000                       // max voxels
#define PC  5                           // max points per voxel
#define FC  4                           // features per point
#define BC  4                           // batch
#define NC  200000                      // points per cloud

#define RMINX (-51.2f)
#define RMINY (-51.2f)
#define RMINZ (-4.0f)
#define VSZ   (0.2f)

#define NBLK ((NC + 255) / 256)         // 782 blocks per batch for per-point kernels
#define ABLK ((BC * NC + 255) / 256)    // 3125 blocks for active-list kernels (worst case)

// output layout (32-bit words, concatenated in return order)
#define VOX_OFF   0                                    // B*V*P*F floats
#define NUM_OFF   (VOX_OFF  + BC * VC * PC * FC)       // B*V int32
#define CRD_OFF   (NUM_OFF  + BC * VC)                 // B*V*4 int32
#define PC_OFF    (CRD_OFF  + BC * VC * 4)             // B*N int32
#define XYZ_OFF   (PC_OFF   + BC * NC)                 // B*V*3 float
#define BCOL_OFF  (XYZ_OFF  + BC * VC * 3)             // B*V int32
#define OUT_TOTAL (BCOL_OFF + BC * VC)                 // 7,760,000 words

// workspace layout (int32 words)
#define CELL_OFF  0                                    // B*G ints (first_occ -> rank -> round scratch)
#define LIN_OFF   (CELL_OFF + (size_t)BC * GC)         // B*N (bit30 = is_first flag, <0 = invalid)
#define SLOT_OFF  (LIN_OFF  + (size_t)BC * NC)         // B*N
#define VID_OFF   (SLOT_OFF + (size_t)BC * NC)         // B*N
#define PART_OFF  (VID_OFF  + (size_t)BC * NC)         // B*N (block-inclusive scan of is_first)
#define BSUM_OFF  (PART_OFF + (size_t)BC * NC)         // B*NBLK
#define CNT_OFF   (BSUM_OFF + (size_t)BC * NBLK)       // 1 (active-list counter)
#define ACT_OFF   (CNT_OFF  + 1)                       // B*N worst case (active point list)
#define FLAG_BIT  (1 << 30)
#define LIN_MASK  (FLAG_BIT - 1)

// ---------------- kernels ----------------

// Initialize all output regions each call (d_out is poisoned by the harness).
__global__ void k_init_out(int* __restrict__ w) {
    int i = blockIdx.x * 256 + threadIdx.x;
    if (i >= OUT_TOTAL) return;
    int v = 0;
    if (i >= BCOL_OFF) {
        v = (i - BCOL_OFF) / VC;                       // batch_col
    } else if (i >= CRD_OFF && i < CRD_OFF + BC * VC * 4) {
        int rel = i - CRD_OFF;                         // voxel_coords: col0=batch, rest 0
        v = ((rel & 3) == 0) ? ((rel >> 2) / VC) : 0;
    }
    w[i] = v;                                          // zero == 0.0f bits for float regions
}

// Fill the per-cell first-occurrence array with sentinel N; zero the active counter.
__global__ void k_fill_cell(int* __restrict__ cell, int* __restrict__ cnt) {
    int i = blockIdx.x * 256 + threadIdx.x;
    if (i == 0) cnt[0] = 0;
    if (i < BC * GC) cell[i] = NC;
}

// Per point: compute cell, validity; atomicMin first-occurrence index.
__global__ void k_bin(const float4* __restrict__ pts, int* __restrict__ lin_arr,
                      int* __restrict__ cell) {
    int i = blockIdx.x * 256 + threadIdx.x;            // per-batch point index
    int b = blockIdx.y;
    if (i >= NC) return;
    int gi = b * NC + i;
    float4 p = pts[gi];
    float fx = floorf((p.x - RMINX) / VSZ);
    float fy = floorf((p.y - RMINY) / VSZ);
    float fz = floorf((p.z - RMINZ) / VSZ);
    bool valid = (fx >= 0.0f) && (fx < (float)GXC) &&
                 (fy >= 0.0f) && (fy < (float)GYC) &&
                 (fz >= 0.0f) && (fz < (float)GZC);
    int lin = -1;
    if (valid) {
        lin = (int)fz * (GXC * GYC) + (int)fy * GXC + (int)fx;
        atomicMin(&cell[(size_t)b * GC + lin], i);
    }
    lin_arr[gi] = lin;
}

// Per point: is_first flag; wave32 ballot scan + per-wave combine; mark bit30 in lin_arr.
__global__ void k_flag_scan(int* __restrict__ lin_arr, const int* __restrict__ cell,
                            int* __restrict__ partial, int* __restrict__ bsum) {
    __shared__ int wsum[8];
    int i = blockIdx.x * 256 + threadIdx.x;
    int b = blockIdx.y;
    int flag = 0;
    if (i < NC) {                                      // no early return: ballot needs all lanes
        int gi = b * NC + i;
        int raw = lin_arr[gi];
        if (raw >= 0 && cell[(size_t)b * GC + raw] == i) {
            flag = 1;
            lin_arr[gi] = raw | FLAG_BIT;
        }
    }
    int lane = threadIdx.x & 31;
    int wv   = threadIdx.x >> 5;
    unsigned bal = (unsigned)__ballot(flag);           // wave32: low 32 bits
    int incl = __popc(bal & (0xFFFFFFFFu >> (31 - lane)));
    if (lane == 31) wsum[wv] = __popc(bal);
    __syncthreads();
    if (threadIdx.x == 0) {
        int acc = 0;
        for (int k = 0; k < 8; k++) { int s = wsum[k]; wsum[k] = acc; acc += s; }
        bsum[b * NBLK + blockIdx.x] = acc;             // block total
    }
    __syncthreads();
    if (i < NC) partial[b * NC + i] = wsum[wv] + incl; // block-inclusive rank count
}

// Exclusive scan of the per-batch block sums (tiny: NBLK=782 per batch).
__global__ void k_scan_bsums(int* __restrict__ bsum) {
    int b = threadIdx.x;
    if (b >= BC) return;
    int acc = 0;
    for (int j = 0; j < NBLK; j++) {
        int s = bsum[b * NBLK + j];
        bsum[b * NBLK + j] = acc;
        acc += s;
    }
}

// First points write their voxel rank into the cell array (reusing it).
__global__ void k_write_rank(const int* __restrict__ lin_arr, const int* __restrict__ partial,
                             const int* __restrict__ bsum, int* __restrict__ cell) {
    int i = blockIdx.x * 256 + threadIdx.x;
    int b = blockIdx.y;
    if (i >= NC) return;
    int gi = b * NC + i;
    int raw = lin_arr[gi];
    if (raw >= 0 && (raw & FLAG_BIT)) {
        int rank = partial[gi] - 1 + bsum[b * NBLK + blockIdx.x];
        cell[(size_t)b * GC + (raw & LIN_MASK)] = rank;
    }
}

__device__ __forceinline__ void place_point(int* w, float* f, int b, int vid, int slot,
                                            float4 p) {
    int row = b * VC + vid;
    float* vx = f + VOX_OFF + ((size_t)row * PC + slot) * FC;
    vx[0] = p.x; vx[1] = p.y; vx[2] = p.z; vx[3] = p.w;
    atomicAdd(&w[NUM_OFF + row], 1);
}

// Per point: rank -> vid; place first points at slot 0; write coords/xyz; baseline pc_voxel_id;
// append unplaced valid-vid points to the compact active list.
__global__ void k_assign(const float4* __restrict__ pts, const int* __restrict__ lin_arr,
                         const int* __restrict__ cell, int* __restrict__ slot_arr,
                         int* __restrict__ vid_arr, int* __restrict__ act,
                         int* __restrict__ cnt, int* __restrict__ w, float* __restrict__ f) {
    int i = blockIdx.x * 256 + threadIdx.x;
    int b = blockIdx.y;
    if (i >= NC) return;
    int gi = b * NC + i;
    int raw = lin_arr[gi];
    int vid = -1, slot = -1;
    if (raw >= 0) {
        int l = raw & LIN_MASK;
        int r = cell[(size_t)b * GC + l];              // rank of this cell's voxel (>= 0)
        if (r < VC) vid = r;
        if ((raw & FLAG_BIT) && vid >= 0) {
            slot = 0;
            float4 p = pts[gi];
            place_point(w, f, b, vid, 0, p);
            int row = b * VC + vid;
            int cz = l / (GXC * GYC);
            int cy = (l >> 9) & (GYC - 1);
            int cx = l & (GXC - 1);
            int* crd = w + CRD_OFF + (size_t)row * 4;
            crd[0] = b; crd[1] = cz; crd[2] = cy; crd[3] = cx;
            float* xo = f + XYZ_OFF + (size_t)row * 3;
            xo[0] = p.x; xo[1] = p.y; xo[2] = p.z;
        } else if (vid >= 0) {
            act[atomicAdd(cnt, 1)] = gi;               // order-free: later ops all commute
        }
    }
    vid_arr[gi]  = vid;
    slot_arr[gi] = slot;
    w[PC_OFF + gi] = (slot == 0) ? vid : -1;           // baseline; rounds overwrite on placement
}

// Round r (r = 1..P-1) over the compact active list: deterministic atomicMin slotting.
__global__ void k_round_reset(const int* __restrict__ act, const int* __restrict__ cnt,
                              const int* __restrict__ lin_arr, const int* __restrict__ slot_arr,
                              int* __restrict__ cell) {
    int j = blockIdx.x * 256 + threadIdx.x;
    if (j >= cnt[0]) return;
    int gi = act[j];
    if (slot_arr[gi] >= 0) return;                     // already placed in an earlier round
    int b = gi / NC;
    cell[(size_t)b * GC + (lin_arr[gi] & LIN_MASK)] = NC;   // sentinel (same value from all)
}

__global__ void k_round_min(const int* __restrict__ act, const int* __restrict__ cnt,
                            const int* __restrict__ lin_arr, const int* __restrict__ slot_arr,
                            int* __restrict__ cell) {
    int j = blockIdx.x * 256 + threadIdx.x;
    if (j >= cnt[0]) return;
    int gi = act[j];
    if (slot_arr[gi] >= 0) return;
    int b = gi / NC;
    atomicMin(&cell[(size_t)b * GC + (lin_arr[gi] & LIN_MASK)], gi - b * NC);
}

__global__ void k_round_resolve(const float4* __restrict__ pts, const int* __restrict__ act,
                                const int* __restrict__ cnt, const int* __restrict__ lin_arr,
                                int* __restrict__ slot_arr, const int* __restrict__ vid_arr,
                                const int* __restrict__ cell, int* __restrict__ w,
                                float* __restrict__ f, int r) {
    int j = blockIdx.x * 256 + threadIdx.x;
    if (j >= cnt[0]) return;
    int gi = act[j];
    if (slot_arr[gi] >= 0) return;
    int b = gi / NC;
    if (cell[(size_t)b * GC + (lin_arr[gi] & LIN_MASK)] == gi - b * NC) {
        int vid = vid_arr[gi];
        slot_arr[gi] = r;
        place_point(w, f, b, vid, r, pts[gi]);
        w[PC_OFF + gi] = vid;                          // finalize pc_voxel_id for placed point
    }
}

// ---------------- launcher ----------------
extern "C" void kernel_launch(void* const* d_in, const int* in_sizes, int n_in,
                              void* d_out, int out_size, void* d_ws, size_t ws_size,
                              hipStream_t stream) {
    const float4* pts = (const float4*)d_in[0];        // [B, N, 4] float32
    int* w   = (int*)d_out;
    float* f = (float*)d_out;
    int* ws  = (int*)d_ws;

    int* cell    = ws + CELL_OFF;
    int* lin_arr = ws + LIN_OFF;
    int* slot_a  = ws + SLOT_OFF;
    int* vid_a   = ws + VID_OFF;
    int* partial = ws + PART_OFF;
    int* bsum    = ws + BSUM_OFF;
    int* cnt     = ws + CNT_OFF;
    int* act     = ws + ACT_OFF;

    dim3 pg(NBLK, BC);

    k_init_out<<<(OUT_TOTAL + 255) / 256, 256, 0, stream>>>(w);
    k_fill_cell<<<((size_t)BC * GC + 255) / 256, 256, 0, stream>>>(cell, cnt);
    k_bin<<<pg, 256, 0, stream>>>(pts, lin_arr, cell);
    k_flag_scan<<<pg, 256, 0, stream>>>(lin_arr, cell, partial, bsum);
    k_scan_bsums<<<1, 64, 0, stream>>>(bsum);
    k_write_rank<<<pg, 256, 0, stream>>>(lin_arr, partial, bsum, cell);
    k_assign<<<pg, 256, 0, stream>>>(pts, lin_arr, cell, slot_a, vid_a, act, cnt, w, f);
    for (int r = 1; r < PC; r++) {
        k_round_reset<<<ABLK, 256, 0, stream>>>(act, cnt, lin_arr, slot_a, cell);
        k_round_min<<<ABLK, 256, 0, stream>>>(act, cnt, lin_arr, slot_a, cell);
        k_round_resolve<<<ABLK, 256, 0, stream>>>(pts, act, cnt, lin_arr, slot_a, vid_a,
                                                  cell, w, f, r);
    }
}